// ResizeCPTransform_65094524339062
// MI455X (gfx1250) — compile-verified
//
#include <hip/hip_runtime.h>
#include <math.h>

// ---------------------------------------------------------------------------
// ResizeCPTransform: depthwise separable 7^3 Gaussian blur (zero-padded SAME)
// fused with trilinear half-pixel resize 160^3 -> 43^3, on 4 batches x 3 ch.
//
// Plan:
//   K0 setup   : 1-D marginals, per-axis fused 8-tap (bilinear (x) gaussian)
//                weights/bases, banded f32 WMMA A-matrices for the x axis.
//   K1 blur_zy : per-(b,zo,yo) 64-tap z/y contraction -> T[4,43,43,160,3]
//                in workspace (14.2 MB, L2-resident on MI455X's 192MB L2).
//   K2 x-GEMM  : OUT = A_band @ T via V_WMMA_F32_16X16X4_F32 (wave32 tiles).
// ---------------------------------------------------------------------------

typedef __attribute__((ext_vector_type(2))) float v2f;
typedef __attribute__((ext_vector_type(8))) float v8f;

#define IN_D   160
#define OUT_D  43
#define NCH    3
#define NB     4
#define KMAX   80                       // padded K band per 16-xo tile (<=64 used)
#define ROWLEN (IN_D * NCH)             // 480
#define NTOT   (NB * OUT_D * OUT_D * NCH) // 22188 N-columns of the x-GEMM
#define NTILE  ((NTOT + 15) / 16)       // 1387
#define TILES  (3 * NTILE)              // 4161 wave-tiles (3 M-tiles)

// workspace layout (float offsets)
#define WS_G1D  0                       // 3*7 marginal kernels (pad to 24)
#define WS_BASE 24                      // 3*43 ints: first tap index per output
#define WS_W8   156                     // 3*43*8 fused tap weights
#define WS_XLO  1188                    // 3 ints: x-band origin per M-tile
#define WS_AW   1192                    // 3*16*80 banded A matrices
#define WS_T    5120                    // 4*43*43*480 intermediate (3,550,080)

__global__ void setup_kernel(const float* __restrict__ kern,
                             float* __restrict__ ws) {
  int t = threadIdx.x;
  int* wsi = (int*)ws;

  // 1) marginals of the normalized separable 7^3 kernel == normalized 1-D kernels
  if (t < 21) {
    int axis = t / 7, i = t % 7;
    float s = 0.f;
    for (int a = 0; a < 7; ++a)
      for (int b = 0; b < 7; ++b) {
        int idx = (axis == 0) ? (i * 49 + a * 7 + b)
                : (axis == 1) ? (a * 49 + i * 7 + b)
                              : (a * 49 + b * 7 + i);
        s += kern[idx];
      }
    ws[WS_G1D + axis * 7 + i] = s;
  }
  __syncthreads();

  // 2) fused 8-tap = (1-f)*g[t] + f*g[t-1], base = floor(center)-3
  if (t < 3 * OUT_D) {
    int axis = t / OUT_D, o = t % OUT_D;
    const float scale = (float)IN_D / (float)OUT_D;
    float center = ((float)o + 0.5f) * scale - 0.5f;  // half-pixel centers
    int i0 = (int)floorf(center);
    float f = center - (float)i0;
    int base = i0 - 3;
    wsi[WS_BASE + axis * OUT_D + o] = base;
    for (int tap = 0; tap < 8; ++tap) {
      float w = 0.f;
      if (tap < 7)  w += (1.f - f) * ws[WS_G1D + axis * 7 + tap];
      if (tap >= 1) w += f * ws[WS_G1D + axis * 7 + tap - 1];
      int x = base + tap;
      if (x < 0 || x >= IN_D) w = 0.f;  // conv SAME zero padding
      ws[WS_W8 + (axis * OUT_D + o) * 8 + tap] = w;
    }
  }
  __syncthreads();

  // 3) banded A matrices for the x-axis GEMM (zero-filled then scattered)
  for (int i = t; i < 3 * 16 * KMAX; i += blockDim.x) ws[WS_AW + i] = 0.f;
  if (t < 3) wsi[WS_XLO + t] = wsi[WS_BASE + 2 * OUT_D + 16 * t];
  __syncthreads();

  if (t < 48) {
    int mt = t / 16, m = t % 16;
    int xo = mt * 16 + m;
    if (xo < OUT_D) {
      int xlo  = wsi[WS_XLO + mt];
      int base = wsi[WS_BASE + 2 * OUT_D + xo];
      for (int tap = 0; tap < 8; ++tap) {
        int x = base + tap;
        if (x >= 0 && x < IN_D)                      // band span < KMAX by design
          ws[WS_AW + (mt * 16 + m) * KMAX + (x - xlo)] =
              ws[WS_W8 + (2 * OUT_D + xo) * 8 + tap];
      }
    }
  }
}

// One workgroup per (b, zo, yo): 64-tap z/y contraction over coalesced rows.
__global__ void blur_zy_kernel(const float* __restrict__ img,
                               float* __restrict__ ws) {
  int blk = blockIdx.x;
  int b  = blk / (OUT_D * OUT_D);
  int rm = blk % (OUT_D * OUT_D);
  int zo = rm / OUT_D, yo = rm % OUT_D;

  const int* wsi = (const int*)ws;
  int zb = wsi[WS_BASE + zo];
  int yb = wsi[WS_BASE + OUT_D + yo];
  float wz[8], wy[8];
#pragma unroll
  for (int i = 0; i < 8; ++i) {
    wz[i] = ws[WS_W8 + zo * 8 + i];
    wy[i] = ws[WS_W8 + (OUT_D + yo) * 8 + i];
  }

  const float* imgB = img + (size_t)b * IN_D * IN_D * ROWLEN;
  float* Trow = ws + WS_T + (size_t)((b * OUT_D + zo) * OUT_D + yo) * ROWLEN;

  for (int v = threadIdx.x; v < ROWLEN; v += blockDim.x) {
    float acc = 0.f;
#pragma unroll
    for (int dz = 0; dz < 8; ++dz) {
      int z = zb + dz;
      if (z < 0 || z >= IN_D) continue;             // zero padding
      const float* pz = imgB + (size_t)z * IN_D * ROWLEN;
      float wzv = wz[dz];
#pragma unroll
      for (int dy = 0; dy < 8; ++dy) {
        int y = yb + dy;
        if (y < 0 || y >= IN_D) continue;
        acc += wzv * wy[dy] * pz[(size_t)y * ROWLEN + v];
      }
    }
    Trow[v] = acc;
  }
}

// x-axis contraction as banded GEMM on WMMA: D[16x16] per wave32 tile.
// A (16xK band, f32): lane&15 = M row, lanes>=16 carry K+2 (ISA 7.12.2).
// B (Kx16, f32): lane&15 = N col, lanes>=16 carry K+2 (mirrored layout).
// D (16x16 f32, 8 VGPRs): vgpr v holds M = v + 8*(lane>=16), N = lane&15.
__global__ void xgemm_wmma_kernel(float* __restrict__ ws,
                                  float* __restrict__ out) {
  int wave = blockIdx.x * (blockDim.x >> 5) + (threadIdx.x >> 5);
  if (wave >= TILES) return;
  int lane = threadIdx.x & 31;
  int mt = wave % 3;
  int nt = wave / 3;

  const int* wsi = (const int*)ws;
  int xlo = wsi[WS_XLO + mt];

  int m    = lane & 15;
  int koff = (lane >> 4) << 1;                      // 0 or 2
  int ncol = nt * 16 + (lane & 15);
  int nc   = ncol < NTOT ? ncol : (NTOT - 1);       // clamp: keep EXEC uniform
  int c  = nc % NCH;
  int r  = nc / NCH;
  int yo = r % OUT_D;
  int zz = r / OUT_D;
  int zo = zz % OUT_D;
  int b  = zz / OUT_D;

  const float* Trow = ws + WS_T +
      (size_t)((b * OUT_D + zo) * OUT_D + yo) * ROWLEN + c;
  const float* Arow = ws + WS_AW + (mt * 16 + m) * KMAX;

  v8f acc = {};
#pragma unroll
  for (int k0 = 0; k0 < KMAX; k0 += 4) {
    v2f a, bf;
    a.x = Arow[k0 + koff];
    a.y = Arow[k0 + koff + 1];
    int x0 = xlo + k0 + koff;                       // clamp: weight is 0 OOB
    int x0c = x0 < 0 ? 0 : (x0 > IN_D - 1 ? IN_D - 1 : x0);
    int x1  = x0 + 1;
    int x1c = x1 < 0 ? 0 : (x1 > IN_D - 1 ? IN_D - 1 : x1);
    bf.x = Trow[x0c * NCH];
    bf.y = Trow[x1c * NCH];
    acc = __builtin_amdgcn_wmma_f32_16x16x4_f32(false, a, false, bf,
                                                (short)0, acc, false, false);
  }

  int mhalf = (lane >> 4) << 3;                     // 0 or 8
  bool nvalid = ncol < NTOT;
#pragma unroll
  for (int v = 0; v < 8; ++v) {
    int xo = mt * 16 + v + mhalf;
    if (nvalid && xo < OUT_D) {
      size_t oidx = (size_t)((b * OUT_D + zo) * OUT_D + yo) * (OUT_D * NCH) +
                    (size_t)xo * NCH + c;
      out[oidx] = acc[v];
    }
  }
}

extern "C" void kernel_launch(void* const* d_in, const int* in_sizes, int n_in,
                              void* d_out, int out_size, void* d_ws, size_t ws_size,
                              hipStream_t stream) {
  (void)in_sizes; (void)n_in; (void)out_size; (void)ws_size;
  const float* img  = (const float*)d_in[0];   // [4,160,160,160,3] f32
  const float* kern = (const float*)d_in[1];   // [7,7,7] f32
  float* out = (float*)d_out;                  // [4,43,43,43,3] f32
  float* ws  = (float*)d_ws;

  hipLaunchKernelGGL(setup_kernel, dim3(1), dim3(256), 0, stream, kern, ws);
  hipLaunchKernelGGL(blur_zy_kernel, dim3(NB * OUT_D * OUT_D), dim3(256), 0,
                     stream, img, ws);
  int blocks = (TILES + 3) / 4;                // 128 threads = 4 wave32 tiles
  hipLaunchKernelGGL(xgemm_wmma_kernel, dim3(blocks), dim3(128), 0, stream,
                     ws, out);
}